// DBNBeatDecoder_40218073759893
// MI455X (gfx1250) — compile-verified
//
#include <hip/hip_runtime.h>
#include <hip/hip_bf16.h>
#include <math.h>

// DBN beat-tracking Viterbi for MI455X (gfx1250).
// One workgroup per batch. Delta vector lives in REGISTERS (22 per thread);
// per step only boundary + block-last values cross threads via double-buffered
// LDS, giving exactly one s_barrier per step. Logit row is DMA'd global->LDS
// via the CDNA5 Tensor Data Mover (TENSORcnt). Max-plus gather uses b128 LDS
// loads from a padded, transposed transition matrix.

#define FRAME_T   6000
#define NBATCH    4
#define NTAU      82        // tempo intervals 28..109
#define MIN_INT   28
#define NSTATE    5617      // sum(28..109)
#define NTHREADS  256       // 8 wave32 waves
#define NPER      22        // 256*22 = 5632 >= 5617; chunk < 28 => <=1 first/last per chunk
#define TROW      84        // transT row stride (82 + 2 sentinel, 16B aligned rows)
#define THRESH    0.05f

typedef __attribute__((ext_vector_type(4))) unsigned int u32x4;
typedef __attribute__((ext_vector_type(8))) int          i32x8;
typedef __attribute__((ext_vector_type(4))) int          i32x4;

__device__ __forceinline__ int firstOf(int k) { return (k * (k + 55)) >> 1; }
__device__ __forceinline__ int lastOf(int k)  { return (((k + 1) * (k + 56)) >> 1) - 1; }

extern "C" __global__ __launch_bounds__(NTHREADS)
void dbn_viterbi_gfx1250(const float* __restrict__ logit,
                         float* __restrict__ out,
                         unsigned char* __restrict__ bp_ws)
{
    __shared__ __attribute__((aligned(16))) float transT[NTAU * TROW]; // [k][i] transposed
    __shared__ __attribute__((aligned(16))) float llog[FRAME_T];       // TDM target
    __shared__ __attribute__((aligned(16))) float lastVals[2][TROW];   // old delta at block-last states
    __shared__ float bnd[2][NTHREADS];                                 // chunk boundary values
    __shared__ float redv[NTHREADS];
    __shared__ int   redi[NTHREADS];
    __shared__ signed char blk[NSTATE];                                // state -> tempo block

    const int tid = threadIdx.x;
    const int b   = blockIdx.x;
    const float*   lg  = logit + (size_t)b * FRAME_T;
    unsigned char* bpb = bp_ws + (size_t)b * (size_t)(FRAME_T - 1) * NTAU;

    // ---- init: transposed transition matrix (float64 math, matches numpy) ----
    if (tid < NTAU) {
        const double taui = (double)(MIN_INT + tid);
        double mx = -1e300;
        for (int j = 0; j < NTAU; ++j) {
            double r = -100.0 * fabs((double)(MIN_INT + j) / taui - 1.0);
            mx = (r > mx) ? r : mx;
        }
        double sum = 0.0;
        for (int j = 0; j < NTAU; ++j) {
            double r = -100.0 * fabs((double)(MIN_INT + j) / taui - 1.0);
            sum += exp(r - mx);
        }
        const double lz = log(sum) + mx;
        for (int j = 0; j < NTAU; ++j) {
            double r = -100.0 * fabs((double)(MIN_INT + j) / taui - 1.0);
            transT[j * TROW + tid] = (float)(r - lz);     // transposed: row = dest tempo j
        }
        transT[tid * TROW + 82] = -1e30f;                 // row sentinels (never win argmax)
        transT[tid * TROW + 83] = -1e30f;
        const int f   = firstOf(tid);
        const int tau = MIN_INT + tid;
        for (int s = 0; s < tau; ++s) blk[f + s] = (signed char)tid;
    }
    if (tid < 2) { lastVals[tid][82] = -1e30f; lastVals[tid][83] = -1e30f; }

    // ---- TDM: DMA 6000-float logit row global->LDS (wave 0 issues once) ----
    if (tid == 0) {
        unsigned long long ga = (unsigned long long)(const void*)lg;
        u32x4 g0;
        g0.x = 1u;                                   // count=1 (valid descriptor)
        g0.y = (unsigned)(size_t)llog;               // lds_addr (flat low 32 = LDS offset)
        g0.z = (unsigned)(ga & 0xFFFFFFFFull);       // global_addr[31:0]
        g0.w = (unsigned)((ga >> 32) & 0x1FFFFFFull) // global_addr[56:32]
             | (2u << 30);                           // type=2 ("image")
        i32x8 g1;
        g1[0] = (2 << 16);                           // data_size=4B
        g1[1] = (int)((FRAME_T & 0xFFFF) << 16);     // tensor_dim0 low16
        g1[2] = (int)(((FRAME_T >> 16) & 0xFFFF) | (1 << 16)); // dim0 hi, tensor_dim1=1
        g1[3] = (int)((FRAME_T & 0xFFFF) << 16);     // tile_dim0 = 6000
        g1[4] = 1;                                   // tile_dim1 = 1
        g1[5] = FRAME_T;                             // tensor_dim0_stride low32
        g1[6] = 0;
        g1[7] = 0;
        i32x4 gz4 = {0, 0, 0, 0};
        i32x8 gz8 = {0, 0, 0, 0, 0, 0, 0, 0};
        __builtin_amdgcn_tensor_load_to_lds(g0, g1, gz4, gz4, gz8, 0);
        __builtin_amdgcn_s_wait_tensorcnt(0);
    }
    __syncthreads();

    // ---- per-thread chunk topology (hoisted out of the 6000-step loop) ----
    const int c = tid * NPER;
    int nval = NSTATE - c;
    nval = (nval < 0) ? 0 : ((nval > NPER) ? NPER : nval);
    int fpos = -1, kF = 0;          // position/tempo of the (unique) first-state in chunk
    int lpos = -1, kL = 0;          // position/tempo of the (unique) last-state in chunk
    for (int j = 0; j < nval; ++j) {
        const int s = c + j;
        const int k = blk[s];
        if (s == firstOf(k)) { fpos = j; kF = k; }
        if (s == lastOf(k))  { lpos = j; kL = k; }
    }

    // ---- t = 0: delta0 = emission - log(S), into registers ----
    float v[NPER];
    float vlast = 0.f;              // mirror of v[lpos] (valid iff lpos >= 0)
    {
        const float x0  = llog[0];
        const float sp0 = fmaxf(x0, 0.f) + log1pf(expf(-fabsf(x0)));
        const float blp0 = x0 - sp0, nblp0 = -sp0;
        const float lgS = logf((float)NSTATE);
#pragma unroll
        for (int j = 0; j < NPER; ++j) {
            v[j] = ((j == fpos) ? blp0 : nblp0) - lgS;
            if (j == lpos) vlast = v[j];
        }
    }

    // ---- forward Viterbi: 6000 steps, ONE barrier per step, regs + tiny LDS ----
    const float4* trowF = (const float4*)&transT[kF * TROW];
    for (int t = 1; t < FRAME_T; ++t) {
        const int buf = t & 1;
        // phase A: publish previous-step values needed across threads
        if (lpos >= 0) lastVals[buf][kL] = vlast;
        bnd[buf][tid] = v[NPER - 1];
        __syncthreads();

        // phase B: compute step t
        const float x     = llog[t];
        const float sp    = fmaxf(x, 0.f) + log1pf(expf(-fabsf(x)));
        const float blpv  = x - sp;     // log sigmoid(x)  -> position-0 states
        const float nblpv = -sp;        // log sigmoid(-x) -> other states
        const float left  = (tid > 0) ? bnd[buf][tid - 1] : -3.4e38f;

        float maxv = -3.4e38f;
        int   bi   = 0;
        if (fpos >= 0) {
            // (max,+) over 82 incoming beat transitions, b128 LDS gathers
            const float4* lvp = (const float4*)lastVals[buf];
            for (int q = 0; q < TROW / 4; ++q) {
                const float4 lv = lvp[q];
                const float4 tr = trowF[q];
                const float c0 = lv.x + tr.x;
                const float c1 = lv.y + tr.y;
                const float c2 = lv.z + tr.z;
                const float c3 = lv.w + tr.w;
                if (c0 > maxv) { maxv = c0; bi = 4 * q + 0; }
                if (c1 > maxv) { maxv = c1; bi = 4 * q + 1; }
                if (c2 > maxv) { maxv = c2; bi = 4 * q + 2; }
                if (c3 > maxv) { maxv = c3; bi = 4 * q + 3; }
            }
            bpb[(size_t)(t - 1) * NTAU + kF] = (unsigned char)bi;
        }

        // phase shift in registers: new[s] = old[s-1] + nblp
#pragma unroll
        for (int j = NPER - 1; j >= 1; --j) v[j] = v[j - 1] + nblpv;
        v[0] = left + nblpv;
#pragma unroll
        for (int j = 0; j < NPER; ++j) {
            if (j == fpos) v[j] = maxv + blpv;   // beat re-entry overrides shift
            if (j == lpos) vlast = v[j];
        }
    }

    // ---- argmax over final delta (first index wins ties, like jnp.argmax) ----
    {
        float bv = -3.4e38f;
        int bidx = NSTATE;
#pragma unroll
        for (int j = 0; j < NPER; ++j) {
            if (j < nval && v[j] > bv) { bv = v[j]; bidx = c + j; }
        }
        redv[tid] = bv;
        redi[tid] = bidx;
        __syncthreads();
        for (int off = NTHREADS / 2; off > 0; off >>= 1) {
            if (tid < off) {
                const float v2 = redv[tid + off];
                const int   i2 = redi[tid + off];
                if (v2 > redv[tid] || (v2 == redv[tid] && i2 < redi[tid])) {
                    redv[tid] = v2; redi[tid] = i2;
                }
            }
            __syncthreads();
        }
    }

    // ---- zero this batch's output row ----
    float* orow = out + (size_t)b * FRAME_T;
    for (int t = tid; t < FRAME_T; t += NTHREADS) orow[t] = 0.f;
    __syncthreads();

    // ---- backtrack: hop beat-to-beat (only position-0 states can emit 1) ----
    if (tid == 0) {
        int s = redi[0];
        int t = FRAME_T - 1;
        for (;;) {
            const int k = blk[s];
            const int f = firstOf(k);
            if (s != f) {
                const int d = s - f;     // phase states implicitly backpoint to s-1
                if (d > t) break;        // chain exits t=0 without hitting a beat
                t -= d;
                s  = f;
            }
            const float act = 1.f / (1.f + expf(-llog[t]));
            if (act >= THRESH) orow[t] = 1.0f;
            if (t == 0) break;
            const int bi = bpb[(size_t)(t - 1) * NTAU + k];
            s = lastOf(bi);              // beat backpointer = last state of tempo bi
            --t;
        }
    }
}

extern "C" void kernel_launch(void* const* d_in, const int* in_sizes, int n_in,
                              void* d_out, int out_size, void* d_ws, size_t ws_size,
                              hipStream_t stream) {
    (void)in_sizes; (void)n_in; (void)out_size; (void)ws_size;
    const float* logit = (const float*)d_in[0];
    float* out = (float*)d_out;
    unsigned char* bp = (unsigned char*)d_ws;   // 4 * 5999 * 82 bytes ~= 1.97 MB

    dbn_viterbi_gfx1250<<<NBATCH, NTHREADS, 0, stream>>>(logit, out, bp);
}